// EnsembleLayer_16020228014825
// MI455X (gfx1250) — compile-verified
//
#include <hip/hip_runtime.h>

// ---------------------------------------------------------------------------
// EnsembleLayer: y[b, n*32+c] = dec_b[n,c] + sum_h dec_w[n,c,h] *
//                 (enc_b[n,h] + sum_k enc_w[n,h,k] * x[b, clusters[n,k]])
// Memory-bound (8 FLOP/byte, ~512 MB traffic -> ~22us floor at 23.3 TB/s).
// fp32 WMMA (v_wmma_f32_16x16x4_f32) chains; gather done via scattered LDS
// writes during fully-coalesced staging of x.
// ---------------------------------------------------------------------------

#define NC 16      // clusters
#define CS 32      // cluster size
#define NH 16      // hidden
#define NF 512     // features = NC*CS
#define ROWS_PER_BLOCK 64
#define THREADS 128            // 4 waves (wave32)
#define XPITCH 516             // 512 + 4 pad dwords -> conflict-free A loads
#define HPITCH 18              // 16 + 2 pad dwords, even for b64 alignment

typedef __attribute__((ext_vector_type(2))) float v2f;
typedef __attribute__((ext_vector_type(4))) float v4f;
typedef __attribute__((ext_vector_type(8))) float v8f;

#if defined(__HIP_DEVICE_COMPILE__)
#if !__has_builtin(__builtin_amdgcn_wmma_f32_16x16x4_f32)
#error "missing __builtin_amdgcn_wmma_f32_16x16x4_f32 on this toolchain"
#endif
#endif

__global__ __launch_bounds__(THREADS) void ensemble_wmma_kernel(
    const float* __restrict__ x,        // [B, 512]
    const int*   __restrict__ clusters, // [16, 32] permutation of 0..511
    const float* __restrict__ enc_w,    // [16, 16, 32]  (n, h, c)
    const float* __restrict__ enc_b,    // [16, 16]
    const float* __restrict__ dec_w,    // [16, 32, 16]  (n, c, h)
    const float* __restrict__ dec_b,    // [16, 32]
    float*       __restrict__ y)        // [B, 512]
{
    extern __shared__ char smem[];
    float* xi    = reinterpret_cast<float*>(smem);                       // 64*516 f32
    int*   inv   = reinterpret_cast<int*>(smem + ROWS_PER_BLOCK*XPITCH*4); // 512 i32
    float* htile = reinterpret_cast<float*>(smem + ROWS_PER_BLOCK*XPITCH*4 + NF*4); // 4*16*18 f32

    const int tid  = threadIdx.x;
    const int row0 = blockIdx.x * ROWS_PER_BLOCK;

    // inverse permutation: LDS slot inv[g] is where global column g lands
    for (int j = tid; j < NF; j += THREADS) inv[clusters[j]] = j;
    __syncthreads();

    // Stage x tile: coalesced 16B non-temporal global reads, scattered 4B LDS
    // writes apply the cluster permutation (xi[r][n*32+c] = x[r][clusters[n][c]]).
    for (int i = tid; i < ROWS_PER_BLOCK * (NF / 4); i += THREADS) {
        const int r  = i >> 7;          // NF/4 == 128 float4 per row
        const int cq = i & 127;
        const v4f v = __builtin_nontemporal_load(
            reinterpret_cast<const v4f*>(x + (size_t)(row0 + r) * NF + cq * 4));
        float* dst = xi + r * XPITCH;
        dst[inv[cq * 4 + 0]] = v.x;
        dst[inv[cq * 4 + 1]] = v.y;
        dst[inv[cq * 4 + 2]] = v.z;
        dst[inv[cq * 4 + 3]] = v.w;
    }
    __syncthreads();

    const int wave = tid >> 5;
    const int lane = tid & 31;
    const int half = lane >> 4;     // 0: lanes 0-15, 1: lanes 16-31
    const int lh   = lane & 15;

    const float* xw = xi + (wave * 16) * XPITCH;   // this wave's 16 batch rows
    float* hw = htile + wave * (NH * HPITCH);      // wave-private h scratch
    const size_t orow0 = (size_t)(row0 + wave * 16) * NF;

    for (int n = 0; n < NC; ++n) {
        // ---- encoder: h(16x16) = xi(16x32) x enc_w^T(32x16), bias in C ----
        v8f acc;
        {
            const float eb = enc_b[n * NH + lh];   // C col == hidden index
            #pragma unroll
            for (int v = 0; v < 8; ++v) acc[v] = eb;
        }
        const float* ew = enc_w + n * NH * CS;     // [h][k]
        #pragma unroll
        for (int t = 0; t < 8; ++t) {
            const int k0 = 4 * t + 2 * half;       // lanes 0-15: K{0,1}; 16-31: K{2,3}
            v2f a = *reinterpret_cast<const v2f*>(xw + lh * XPITCH + n * CS + k0);
            v2f b = *reinterpret_cast<const v2f*>(ew + lh * CS + k0); // B[k][h]=enc_w[h][k]
            acc = __builtin_amdgcn_wmma_f32_16x16x4_f32(
                false, a, false, b, (short)0, acc, false, false);
        }

        // ---- D-layout -> row-major h[row][h] (wave-private, no barrier) ----
        #pragma unroll
        for (int v = 0; v < 8; ++v)
            hw[(v + 8 * half) * HPITCH + lh] = acc[v];

        // ---- decoder A-frags (reused for both 16-col output tiles) ----
        v2f ha[4];
        #pragma unroll
        for (int t = 0; t < 4; ++t)
            ha[t] = *reinterpret_cast<const v2f*>(hw + lh * HPITCH + 4 * t + 2 * half);

        const float* dwp = dec_w + n * CS * NH;    // [c][h]
        #pragma unroll
        for (int ct = 0; ct < 2; ++ct) {
            v8f dacc;
            {
                const float db = dec_b[n * CS + ct * 16 + lh];
                #pragma unroll
                for (int v = 0; v < 8; ++v) dacc[v] = db;
            }
            #pragma unroll
            for (int t = 0; t < 4; ++t) {
                const int k0 = 4 * t + 2 * half;
                v2f b = *reinterpret_cast<const v2f*>(
                    dwp + (ct * 16 + lh) * NH + k0);  // B[k][c]=dec_w[c][k]
                dacc = __builtin_amdgcn_wmma_f32_16x16x4_f32(
                    false, ha[t], false, b, (short)0, dacc, false, false);
            }
            // ---- store D: lane holds col lh of rows v(+8), non-temporal ----
            #pragma unroll
            for (int v = 0; v < 8; ++v) {
                const int r = v + 8 * half;
                __builtin_nontemporal_store(
                    dacc[v], y + orow0 + (size_t)r * NF + n * CS + ct * 16 + lh);
            }
        }
    }
}

extern "C" void kernel_launch(void* const* d_in, const int* in_sizes, int n_in,
                              void* d_out, int out_size, void* d_ws, size_t ws_size,
                              hipStream_t stream) {
    const float* x        = (const float*)d_in[0];
    const int*   clusters = (const int*)d_in[1];
    const float* enc_w    = (const float*)d_in[2];
    const float* enc_b    = (const float*)d_in[3];
    const float* dec_w    = (const float*)d_in[4];
    const float* dec_b    = (const float*)d_in[5];
    float* y = (float*)d_out;

    const int batch = in_sizes[0] / NF;
    const int grid  = batch / ROWS_PER_BLOCK;   // 131072/64 = 2048

    const size_t shmem = (size_t)ROWS_PER_BLOCK * XPITCH * 4   // xi tile
                       + (size_t)NF * 4                        // inv perm
                       + (size_t)4 * NH * HPITCH * 4;          // h scratch (4 waves)

    ensemble_wmma_kernel<<<dim3(grid), dim3(THREADS), shmem, stream>>>(
        x, clusters, enc_w, enc_b, dec_w, dec_b, y);
}